// HumanLikeWorkingMemory_20985210208346
// MI455X (gfx1250) — compile-verified
//
#include <hip/hip_runtime.h>
#include <math.h>

// ---------------------------------------------------------------------------
// HumanLikeWorkingMemory for MI455X (gfx1250)
//
// Phase 1 (parallel, matrix pipe): P[b][n] = tanh( sum_k X[b][k]*W_add[n][k] + b_add[n] )
//   via V_WMMA_F32_16X16X4_F32 (fp32-exact, GEMM is far from the bottleneck).
//   impAll[b] = sigmoid( W_imp . X[b] + b_imp )  (wave-per-row dot).
// Phase 2 (serial): single workgroup scan over 32768 steps; slots live in
//   registers (thread d owns dim d of all 7 slots), Gram/norm reductions via
//   wave32 shfl_xor butterflies + LDS cross-wave combine, noise stream hidden
//   with global_prefetch.
// ---------------------------------------------------------------------------

typedef float v2f __attribute__((ext_vector_type(2)));
typedef float v8f __attribute__((ext_vector_type(8)));

#define DM    256
#define CAP   7
#define BATCH 32768
#define NPAIR 28                       // pairs (i<=j) over 7 slots
#define PIDX(i, j) ((i) * CAP - ((i) * ((i) - 1)) / 2 + ((j) - (i)))

// ---------------- Phase 1a: GEMM + bias + tanh via f32 WMMA -----------------
// One wave computes one 16x16 tile of P. M tiles = 2048, N tiles = 16.
__global__ __launch_bounds__(256) void hlwm_add_gemm(
    const float* __restrict__ X,      // [32768,256]
    const float* __restrict__ Wadd,   // [256,256] row-major: Wadd[n][k]
    const float* __restrict__ badd,   // [256]
    float* __restrict__ P)            // [32768,256]
{
  const int lane = threadIdx.x & 31;
  const int wv   = threadIdx.x >> 5;
  const int tile = blockIdx.x * 8 + wv;       // 4096 blocks * 8 waves = 32768 tiles
  const int mt   = tile >> 4;                 // 0..2047
  const int nt   = tile & 15;                 // 0..15
  const int half = (lane & 16) ? 1 : 0;       // upper lane half
  const int ko   = half * 2;                  // K sub-offset per ISA A/B layout

  const float* __restrict__ pA = X    + (size_t)(mt * 16 + (lane & 15)) * DM + ko;
  const float* __restrict__ pB = Wadd + (size_t)(nt * 16 + (lane & 15)) * DM + ko;

  v8f acc = {};
  #pragma unroll 8
  for (int k = 0; k < DM; k += 4) {
    v2f a, b;
    a.x = pA[k]; a.y = pA[k + 1];             // A 16x4: lanes<16 K={0,1}, lanes>=16 K={2,3}
    b.x = pB[k]; b.y = pB[k + 1];             // B 4x16: same K striping, N = lane&15
    acc = __builtin_amdgcn_wmma_f32_16x16x4_f32(
        false, a, false, b, (short)0, acc, false, false);
  }

  const int n    = nt * 16 + (lane & 15);
  const int m0   = mt * 16 + half * 8;        // C/D: VGPR r -> M=r (+8 for upper lanes)
  const float bias = badd[n];
  #pragma unroll
  for (int r = 0; r < 8; ++r)
    P[(size_t)(m0 + r) * DM + n] = tanhf(acc[r] + bias);
}

// ---------------- Phase 1b: importance = sigmoid(W_imp . x_b) ---------------
__global__ __launch_bounds__(256) void hlwm_importance(
    const float* __restrict__ X, const float* __restrict__ Wimp,
    const float* __restrict__ bimp, float* __restrict__ impAll)
{
  const int lane = threadIdx.x & 31;
  const int wv   = threadIdx.x >> 5;
  const int row  = blockIdx.x * 8 + wv;       // 4096 blocks * 8 waves = 32768 rows
  const float* __restrict__ xr = X + (size_t)row * DM;
  float s = 0.f;
  #pragma unroll
  for (int i = 0; i < 8; ++i) {
    const int d = lane + i * 32;
    s += xr[d] * Wimp[d];
  }
  #pragma unroll
  for (int off = 16; off > 0; off >>= 1) s += __shfl_xor(s, off, 32);
  if (lane == 0) {
    const float z = s + bimp[0];
    impAll[row] = 1.f / (1.f + expf(-z));
  }
}

// ---------------- Phase 2: sequential slot-dynamics scan --------------------
// Single block, 256 threads; thread d owns dim d of all 7 slots. Uniform state
// (ages, per-slot importance, occupancy bitmask) is replicated per thread.
// P is consumed in place from d_out: row b is read, then overwritten with out[b].
__global__ __launch_bounds__(256) void hlwm_scan(
    float* __restrict__ PO,                   // [32768,256] in: tanh-add, out: result
    const float* __restrict__ impAll,         // [32768]
    const float* __restrict__ noise,          // [32768,7,256]
    const float* __restrict__ slots_init)     // [7,256]
{
  const int d    = threadIdx.x;
  const int lane = d & 31;
  const int wv   = d >> 5;

  __shared__ float wred[8][32];               // per-wave partial sums (28 / 7 used)
  __shared__ float G[32];                     // reduced Gram / norm^2 values

  float s[CAP], ages[CAP], impS[CAP];
  #pragma unroll
  for (int i = 0; i < CAP; ++i) {
    s[i]    = slots_init[i * DM + d];
    ages[i] = 0.f;
    impS[i] = 0.f;
  }
  unsigned occ = 0u;

  for (int b = 0; b < BATCH; ++b) {
    // Hide the noise stream (7 KB/step) and the add row behind the step.
    if (b + 4 < BATCH)
      __builtin_prefetch(noise + (size_t)(b + 4) * (CAP * DM) + d * 7, 0, 3);
    if (d < 32 && b + 2 < BATCH)
      __builtin_prefetch(PO + (size_t)(b + 2) * DM + d * 8, 0, 3);

    const float importance = impAll[b];
    const float add = PO[(size_t)b * DM + d];

    // ---- slot selection (uniform) ----
    const bool has_empty = (occ != 0x7Fu);
    int slot_idx = 0;
    if (has_empty) {
      while ((occ >> slot_idx) & 1u) ++slot_idx;           // first empty
    } else {
      float m = impS[0];
      #pragma unroll
      for (int i = 1; i < CAP; ++i)
        if (impS[i] < m) { m = impS[i]; slot_idx = i; }    // first least-important
    }
    const bool should = has_empty || (importance > 0.3f);
    const unsigned hit = should ? (1u << slot_idx) : 0u;
    const float newc = add * importance;
    #pragma unroll
    for (int i = 0; i < CAP; ++i) {
      if ((hit >> i) & 1u) { s[i] = newc; impS[i] = importance; ages[i] = 0.f; }
      ages[i] += 1.f;
    }
    occ |= hit;
    const bool multi = (__popc(occ) > 1);                  // uniform branch

    if (multi) {
      // ---- Gram S.S^T: 28 pair reductions over 256 dims ----
      float part[NPAIR];
      {
        int p = 0;
        #pragma unroll
        for (int i = 0; i < CAP; ++i)
          #pragma unroll
          for (int j = i; j < CAP; ++j)
            part[p++] = s[i] * s[j];
      }
      #pragma unroll
      for (int p = 0; p < NPAIR; ++p) {
        float v = part[p];
        #pragma unroll
        for (int off = 16; off > 0; off >>= 1) v += __shfl_xor(v, off, 32);
        if (lane == 0) wred[wv][p] = v;
      }
      __syncthreads();
      if (d < NPAIR) {
        float t = 0.f;
        #pragma unroll
        for (int w = 0; w < 8; ++w) t += wred[w][d];
        G[d] = t;
      }
      __syncthreads();

      // ---- sims, interference counts (uniform) ----
      float rinv[CAP];
      #pragma unroll
      for (int i = 0; i < CAP; ++i)
        rinv[i] = 1.f / fmaxf(sqrtf(G[PIDX(i, i)]), 1e-8f);
      float cnt[CAP];
      #pragma unroll
      for (int i = 0; i < CAP; ++i) {
        float c = 0.f;
        #pragma unroll
        for (int j = 0; j < CAP; ++j) {
          const int lo = (i < j) ? i : j, hi = (i < j) ? j : i;
          const float sim = G[PIDX(lo, hi)] * rinv[i] * rinv[j];
          const bool it = (sim > 0.8f) && (sim < 1.0f) &&
                          ((occ >> i) & 1u) && ((occ >> j) & 1u);
          c += it ? 1.f : 0.f;
        }
        cnt[i] = c;
      }

      // ---- noise injection + age decay ----
      #pragma unroll
      for (int i = 0; i < CAP; ++i) {
        float sv = s[i];
        if (cnt[i] != 0.f)
          sv += noise[(size_t)b * (CAP * DM) + i * DM + d] * 0.05f * cnt[i];
        const float dec = ((occ >> i) & 1u) ? expf(-0.005f * ages[i]) : 1.f;
        s[i] = sv * dec;
      }

      // ---- weak-slot cleanup: needs post-update norms (7 reductions) ----
      __syncthreads();                                     // wred/G reuse safety
      #pragma unroll
      for (int i = 0; i < CAP; ++i) {
        float v = s[i] * s[i];
        #pragma unroll
        for (int off = 16; off > 0; off >>= 1) v += __shfl_xor(v, off, 32);
        if (lane == 0) wred[wv][i] = v;
      }
      __syncthreads();
      if (d < CAP) {
        float t = 0.f;
        #pragma unroll
        for (int w = 0; w < 8; ++w) t += wred[w][d];
        G[d] = t;
      }
      __syncthreads();
      #pragma unroll
      for (int i = 0; i < CAP; ++i) {
        const bool weak = ((occ >> i) & 1u) && (sqrtf(G[i]) < 0.05f);
        if (weak) { occ &= ~(1u << i); s[i] = 0.f; impS[i] = 0.f; }
      }
      __syncthreads();                                     // protect G vs next step
    }

    // ---- readout: importance-weighted sum of occupied slots ----
    float wsum = 0.f;
    #pragma unroll
    for (int i = 0; i < CAP; ++i)
      if ((occ >> i) & 1u) wsum += impS[i];
    const float inv = 1.f / (wsum + 1e-8f);
    float o = 0.f;
    #pragma unroll
    for (int i = 0; i < CAP; ++i)
      if ((occ >> i) & 1u) o += (impS[i] * inv) * s[i];
    PO[(size_t)b * DM + d] = o;                            // overwrite consumed row
  }
}

// ---------------------------------------------------------------------------
extern "C" void kernel_launch(void* const* d_in, const int* in_sizes, int n_in,
                              void* d_out, int out_size, void* d_ws, size_t ws_size,
                              hipStream_t stream) {
  const float* x          = (const float*)d_in[0];   // [32768,256]
  const float* noise      = (const float*)d_in[1];   // [32768,7,256]
  const float* slots_init = (const float*)d_in[2];   // [7,256]
  // d_in[3] W_erase, d_in[4] b_erase: computed-but-unused in reference
  const float* W_add      = (const float*)d_in[5];   // [256,256]
  const float* b_add      = (const float*)d_in[6];   // [256]
  const float* W_imp      = (const float*)d_in[7];   // [1,256]
  const float* b_imp      = (const float*)d_in[8];   // [1]

  float* out    = (float*)d_out;                     // [32768,256]; also holds P in-place
  float* impAll = (float*)d_ws;                      // [32768] (128 KB scratch)

  // Phase 1: batched GEMM (WMMA f32) + importance dots.
  hlwm_add_gemm<<<dim3(BATCH * (DM / 16) / (16 * 8)), dim3(256), 0, stream>>>(
      x, W_add, b_add, out);
  hlwm_importance<<<dim3(BATCH / 8), dim3(256), 0, stream>>>(
      x, W_imp, b_imp, impAll);

  // Phase 2: serial scan (single workgroup).
  hlwm_scan<<<dim3(1), dim3(256), 0, stream>>>(out, impAll, noise, slots_init);
}